// LSTM_69724499083456
// MI455X (gfx1250) — compile-verified
//
#include <hip/hip_runtime.h>
#include <hip/hip_bf16.h>

#define T_SEQ  512
#define NBATCH 256
#define NIN    64
#define NHID   256

typedef __attribute__((ext_vector_type(2))) float v2f;
typedef __attribute__((ext_vector_type(8))) float v8f;

__device__ __forceinline__ float sigmoid_f(float x) {
    return 1.0f / (1.0f + __expf(-x));
}
__device__ __forceinline__ float tanh_f(float x) {
#if __has_builtin(__builtin_amdgcn_tanhf)
    return __builtin_amdgcn_tanhf(x);      // v_tanh_f32 (gfx1250 TRANS op)
#else
    // 2*sigmoid(2x)-1 : saturates cleanly to +/-1 for large |x|, no inf/inf
    return 2.0f / (1.0f + __expf(-2.0f * x)) - 1.0f;
#endif
}

__device__ __forceinline__ v8f wmma_f32(v2f a, v2f b, v8f c) {
    // D = A(16x4,f32) * B(4x16,f32) + C(16x16,f32)
    return __builtin_amdgcn_wmma_f32_16x16x4_f32(
        /*neg_a=*/false, a, /*neg_b=*/false, b,
        /*c_mod=*/(short)0, c, /*reuse_a=*/false, /*reuse_b=*/false);
}

// Uniform base pointer (SGPR pair) + 32-bit per-lane element offset
// -> backend emits global_load_b64 saddr form, k0 folds into imm offset.
__device__ __forceinline__ v2f gload2(const float* __restrict__ base, unsigned idx) {
    return *(const v2f*)(base + idx);
}

// One workgroup owns a 16-row batch tile for the whole 512-step scan.
// 16 waves; wave w owns hidden columns [16w, 16w+16).
// h (16 x 256) shared between steps via LDS; c kept in accumulator regs.
__global__ __launch_bounds__(512, 1) void lstm_fused_kernel(
    const float* __restrict__ x,
    const float* __restrict__ Wf, const float* __restrict__ bWf,
    const float* __restrict__ Uf, const float* __restrict__ bUf,
    const float* __restrict__ Wi, const float* __restrict__ bWi,
    const float* __restrict__ Ui, const float* __restrict__ bUi,
    const float* __restrict__ Wo, const float* __restrict__ bWo,
    const float* __restrict__ Uo, const float* __restrict__ bUo,
    const float* __restrict__ Wc, const float* __restrict__ bWc,
    const float* __restrict__ Uc, const float* __restrict__ bUc,
    float* __restrict__ out)
{
    __shared__ float h_lds[16 * NHID];   // 16 KB: h_{t-1} for this batch tile

    const int lane = (int)(threadIdx.x & 31u);
    const int wave = (int)(threadIdx.x >> 5);   // 0..15
    const int m16  = lane & 15;                 // A-row / B-col / D-col index
    const int g2   = lane >> 4;                 // half-wave select (K pair / M+8)
    const int ncol = wave * 16 + m16;           // this lane's global hidden column
    const int b0   = (int)blockIdx.x * 16;      // batch tile base

    // h0 = 0
    for (int idx = (int)threadIdx.x; idx < 16 * NHID; idx += (int)blockDim.x)
        h_lds[idx] = 0.0f;
    __syncthreads();

    // biases depend only on the output column -> fold bW+bU once per lane
    const float bfs = bWf[ncol] + bUf[ncol];
    const float bis = bWi[ncol] + bUi[ncol];
    const float bos = bWo[ncol] + bUo[ncol];
    const float bcs = bWc[ncol] + bUc[ncol];

    // 32-bit per-lane element offsets into the (uniform-base) weight arrays.
    // B-fragment element: B[k][n] = U[ncol][k] pairs (k0+2g2, k0+2g2+1).
    const unsigned offW = (unsigned)ncol * NIN  + 2u * (unsigned)g2;
    const unsigned offU = (unsigned)ncol * NHID + 2u * (unsigned)g2;

    // cell state for this wave's 16x16 tile, in C/D accumulator layout:
    // element r of v8f <-> (row m = r + 8*g2, col ncol)
    v8f cSt = {0.f, 0.f, 0.f, 0.f, 0.f, 0.f, 0.f, 0.f};

    float* out_hseq = out;                                     // (T,B,H)
    float* out_h    = out + (size_t)T_SEQ * NBATCH * NHID;     // (1,B,H)
    float* out_c    = out_h + (size_t)NBATCH * NHID;           // (1,B,H)

    // A-fragment LDS row base for this lane (row m16, element pair at 2*g2)
    const float* hrow = h_lds + m16 * NHID + 2 * g2;

    for (int t = 0; t < T_SEQ; ++t) {
        v8f aF = {0.f,0.f,0.f,0.f,0.f,0.f,0.f,0.f};
        v8f aI = {0.f,0.f,0.f,0.f,0.f,0.f,0.f,0.f};
        v8f aO = {0.f,0.f,0.f,0.f,0.f,0.f,0.f,0.f};
        v8f aC = {0.f,0.f,0.f,0.f,0.f,0.f,0.f,0.f};

        // ---- input projection slice: A = x[t, b0+m, k]  (K = 64) ----
        const unsigned offX =
            (unsigned)(((size_t)t * NBATCH + b0 + m16) * NIN) + 2u * (unsigned)g2;
#pragma unroll
        for (int k0 = 0; k0 < NIN; k0 += 4) {
            v2f a = gload2(x, offX + (unsigned)k0);
            aF = wmma_f32(a, gload2(Wf, offW + (unsigned)k0), aF);
            aI = wmma_f32(a, gload2(Wi, offW + (unsigned)k0), aI);
            aO = wmma_f32(a, gload2(Wo, offW + (unsigned)k0), aO);
            aC = wmma_f32(a, gload2(Wc, offW + (unsigned)k0), aC);
        }

        // ---- recurrent slice: A = h[m, k] from LDS  (K = 256) ----
#pragma unroll 8
        for (int k0 = 0; k0 < NHID; k0 += 4) {
            v2f a = *(const v2f*)(hrow + k0);
            aF = wmma_f32(a, gload2(Uf, offU + (unsigned)k0), aF);
            aI = wmma_f32(a, gload2(Ui, offU + (unsigned)k0), aI);
            aO = wmma_f32(a, gload2(Uo, offU + (unsigned)k0), aO);
            aC = wmma_f32(a, gload2(Uc, offU + (unsigned)k0), aC);
        }

        // ---- gate nonlinearities + cell/hidden update (per-lane, 8 rows) ----
        v8f hN;
#pragma unroll
        for (int r = 0; r < 8; ++r) {
            float f = sigmoid_f(aF[r] + bfs);
            float i = sigmoid_f(aI[r] + bis);
            float o = sigmoid_f(aO[r] + bos);
            float g = tanh_f(aC[r] + bcs);
            float cc = f * cSt[r] + i * g;
            cSt[r] = cc;
            hN[r] = o * tanh_f(cc);
        }

        // global h_seq stores don't need the barrier -> issue them first
#pragma unroll
        for (int r = 0; r < 8; ++r) {
            const int mrow = r + 8 * g2;
            out_hseq[((size_t)t * NBATCH + b0 + mrow) * NHID + ncol] = hN[r];
        }

        __syncthreads();   // all waves done reading h_{t-1}
#pragma unroll
        for (int r = 0; r < 8; ++r) {
            const int mrow = r + 8 * g2;
            h_lds[mrow * NHID + ncol] = hN[r];
        }
        __syncthreads();   // h_t visible to every wave for step t+1
    }

    // final h, c
#pragma unroll
    for (int r = 0; r < 8; ++r) {
        const int mrow = r + 8 * g2;
        out_h[(size_t)(b0 + mrow) * NHID + ncol] = h_lds[mrow * NHID + ncol];
        out_c[(size_t)(b0 + mrow) * NHID + ncol] = cSt[r];
    }
}

extern "C" void kernel_launch(void* const* d_in, const int* in_sizes, int n_in,
                              void* d_out, int out_size, void* d_ws, size_t ws_size,
                              hipStream_t stream) {
    (void)in_sizes; (void)n_in; (void)out_size; (void)d_ws; (void)ws_size;
    const float* x   = (const float*)d_in[0];
    const float* Wf  = (const float*)d_in[1];
    const float* bWf = (const float*)d_in[2];
    const float* Uf  = (const float*)d_in[3];
    const float* bUf = (const float*)d_in[4];
    const float* Wi  = (const float*)d_in[5];
    const float* bWi = (const float*)d_in[6];
    const float* Ui  = (const float*)d_in[7];
    const float* bUi = (const float*)d_in[8];
    const float* Wo  = (const float*)d_in[9];
    const float* bWo = (const float*)d_in[10];
    const float* Uo  = (const float*)d_in[11];
    const float* bUo = (const float*)d_in[12];
    const float* Wc  = (const float*)d_in[13];
    const float* bWc = (const float*)d_in[14];
    const float* Uc  = (const float*)d_in[15];
    const float* bUc = (const float*)d_in[16];

    lstm_fused_kernel<<<dim3(NBATCH / 16), dim3(512), 0, stream>>>(
        x, Wf, bWf, Uf, bUf, Wi, bWi, Ui, bUi,
        Wo, bWo, Uo, bUo, Wc, bWc, Uc, bUc, (float*)d_out);
}